// InvGridSamplerNumerator_1580547968540
// MI455X (gfx1250) — compile-verified
//
#include <hip/hip_runtime.h>
#include <hip/hip_bf16.h>

// Problem constants (fixed by the reference: x is (8,16,256,256) f32,
// inv_grid is (8,256,256,2) f32, output is (8,16,256,256) f32).
#define B_ 8
#define C_ 16
#define H_ 256
#define W_ 256
#define HW_ (H_ * W_)           // 65536
#define CHW_ (C_ * HW_)         // per-batch plane stride (1M floats)
#define NPTS_ (B_ * HW_)        // 524288 scatter points
#define NOUT_ (B_ * C_ * HW_)   // 8388608 output floats

// Non-returning GLOBAL_ATOMIC_ADD_F32 in saddr (GVS) form:
//   mem_addr = SGPR_base + VGPR_offset(i32) + IOFFSET(i24)
// The scalar base is wave-uniform (derived from blockIdx), the per-lane VGPR
// offset is the scatter target byte offset, and the channel stride is folded
// into the 24-bit immediate (c*HW*4 <= 3.9MB < 8MB). No per-atomic VALU
// address math, no return value (STOREcnt-tracked), no CAS loop.
__device__ __forceinline__ void gatomic_add_f32_s(const float* sbase, int voff_bytes,
                                                  float val, int imm_bytes_tag);

#define GATOMIC_ADD_F32_S(sbase, voff, v, imm)                                 \
    asm volatile("global_atomic_add_f32 %0, %1, %2 offset:%c3"                 \
                 :                                                             \
                 : "v"(voff), "v"(v), "s"(sbase), "i"(imm))

__global__ __launch_bounds__(256) void igs_zero_kernel(float4* __restrict__ out) {
    int i = blockIdx.x * 256 + threadIdx.x;
    if (i < (NOUT_ / 4)) {
        out[i] = make_float4(0.0f, 0.0f, 0.0f, 0.0f);
    }
}

__global__ __launch_bounds__(256) void igs_scatter_kernel(
    const float* __restrict__ x,        // (B,C,H,W)
    const float* __restrict__ inv_grid, // (B,H,W,2)
    float* __restrict__ out)            // (B,C,H,W), pre-zeroed
{
    const int bid = blockIdx.x;          // 2048 blocks; 256 blocks per batch
    const int tid = threadIdx.x;
    const int idx = (bid << 8) | tid;    // global point id in [0, NPTS_)
    const int b   = bid >> 8;            // batch index: wave-uniform (SGPR)
    const int hw  = ((bid & 255) << 8) | tid;  // per-lane h*W+w

    // Coalesced 64-bit load of (gi, gj) per lane.
    const float2 g = reinterpret_cast<const float2*>(inv_grid)[idx];

    // Reference math, f32 throughout. Upper clip bound 257 - 2e-10 == 257.0f.
    float ci = ((g.x + 1.0f) * 0.5f) * (float)H_ + 1.0f;
    float cj = ((g.y + 1.0f) * 0.5f) * (float)W_ + 1.0f;
    ci = fminf(fmaxf(ci, 0.0f), 257.0f);
    cj = fminf(fmaxf(cj, 0.0f), 257.0f);

    const int i0 = (int)floorf(ci);
    const int j0 = (int)floorf(cj);

    // Bilinear weights exactly as reference: relu(1 - |c - (c0 + d)|).
    const float wi0 = fmaxf(1.0f - fabsf(ci - (float)i0),       0.0f);
    const float wi1 = fmaxf(1.0f - fabsf(ci - (float)(i0 + 1)), 0.0f);
    const float wj0 = fmaxf(1.0f - fabsf(cj - (float)j0),       0.0f);
    const float wj1 = fmaxf(1.0f - fabsf(cj - (float)(j0 + 1)), 0.0f);

    // Output indices after cropping the 1-wide pad: oi = (i0 + di) - 1.
    const int oi0 = i0 - 1, oi1 = i0;
    const int oj0 = j0 - 1, oj1 = j0;
    const bool vi0 = ((unsigned)oi0 < (unsigned)H_);
    const bool vi1 = ((unsigned)oi1 < (unsigned)H_);
    const bool vj0 = ((unsigned)oj0 < (unsigned)W_);
    const bool vj1 = ((unsigned)oj1 < (unsigned)W_);

    int   off[4];
    float wgt[4];
    bool  val[4];
    off[0] = oi0 * W_ + oj0;  wgt[0] = wi0 * wj0;  val[0] = vi0 && vj0;
    off[1] = oi0 * W_ + oj1;  wgt[1] = wi0 * wj1;  val[1] = vi0 && vj1;
    off[2] = oi1 * W_ + oj0;  wgt[2] = wi1 * wj0;  val[2] = vi1 && vj0;
    off[3] = oi1 * W_ + oj1;  wgt[3] = wi1 * wj1;  val[3] = vi1 && vj1;

    // Scalar (wave-uniform) bases for this batch.
    const float* xb   = x   + (size_t)b * CHW_ + hw;  // per-lane hw folded in
    const float* outb = out + (size_t)b * CHW_;       // stays in SGPR pair

    // Issue all 16 channel loads up front (coalesced b32 per wave32, scalar
    // base + immediate channel offsets) so the memory system stacks latency.
    float xv[C_];
#pragma unroll
    for (int c = 0; c < C_; ++c) {
        xv[c] = xb[(size_t)c * HW_];
    }

    // k-outer / c-inner: one exec-mask guard per corner covers 16 atomics.
    // val[k] is per-lane but constant across channels; statistically it is
    // false only ~1/256 of the time (crop boundary).
#pragma unroll
    for (int k = 0; k < 4; ++k) {
        if (val[k]) {
            const int vo = off[k] * 4;   // per-lane byte offset within plane
            const float wk = wgt[k];
#pragma unroll
            for (int c = 0; c < C_; ++c) {
                const float v = xv[c] * wk;
                GATOMIC_ADD_F32_S(outb, vo, v, c * HW_ * 4);
            }
        }
    }
}

extern "C" void kernel_launch(void* const* d_in, const int* in_sizes, int n_in,
                              void* d_out, int out_size, void* d_ws, size_t ws_size,
                              hipStream_t stream) {
    (void)in_sizes; (void)n_in; (void)out_size; (void)d_ws; (void)ws_size;

    const float* x        = (const float*)d_in[0];
    const float* inv_grid = (const float*)d_in[1];
    float*       out      = (float*)d_out;

    // 1) Zero the accumulation target (global_store_b128, 2M threads).
    {
        const int n4 = NOUT_ / 4;
        const int blocks = (n4 + 255) / 256;
        igs_zero_kernel<<<blocks, 256, 0, stream>>>((float4*)out);
    }
    // 2) Scatter-accumulate: one thread per (b,h,w) source point.
    {
        const int blocks = NPTS_ / 256;
        igs_scatter_kernel<<<blocks, 256, 0, stream>>>(x, inv_grid, out);
    }
}